// ShortConvolution_54941221651145
// MI455X (gfx1250) — compile-verified
//
#include <hip/hip_runtime.h>

// Depthwise causal conv (K=4) + SiLU, fp32. Shape fixed by reference:
// B=4, T=4096, D=2048. Memory-bound: ~268 MB traffic -> ~11.5 us floor @ 23.3 TB/s.
// WMMA inapplicable (depthwise = block-diagonal, no shared-operand GEMM structure).
// Target: pure HBM streaming — b128 coalesced loads/stores, NT cache hints,
// global_prefetch_b8, register sliding window (x read exactly once), wave32 occupancy.
// All dims constexpr so index math is shifts + immediate vmem offsets (no int-div).

typedef float v4f __attribute__((ext_vector_type(4)));

namespace cfg {
constexpr int B  = 4;
constexpr int T  = 4096;
constexpr int D  = 2048;
constexpr int TT = 16;            // time steps per thread
constexpr int Dv = D / 4;         // 512 float4 columns
constexpr int tilesT = T / TT;    // 256
constexpr int DvShift = 9;        // log2(Dv)
constexpr int ttShift = 8;        // log2(tilesT)
}  // namespace cfg

__global__ __launch_bounds__(256) void dwconv4_silu_kernel(
    const float* __restrict__ x,     // (B, T, D) d fastest
    const float* __restrict__ w,     // (D, 4)    k fastest
    float* __restrict__ out)         // (B, T, D)
{
    using namespace cfg;

    const int gid = blockIdx.x * blockDim.x + threadIdx.x;
    const int d4  = gid & (Dv - 1);                       // lane-fast -> coalesced 512B/wave
    const int tt  = (gid >> DvShift) & (tilesT - 1);      // block-uniform
    const int b   = gid >> (DvShift + ttShift);           // block-uniform

    const int t0 = tt * TT;

    // Base pointer for this thread's tile; all accesses become base + imm offset.
    const v4f* __restrict__ xin  =
        (const v4f*)(x + (size_t)b * T * D) + d4 + (size_t)t0 * Dv;
    v4f* __restrict__ yout =
        (v4f*)(out + (size_t)b * T * D) + d4 + (size_t)t0 * Dv;
    const v4f* __restrict__ wv = (const v4f*)w + d4 * 4;

    // gfx1250 prefetch path (global_prefetch_b8): kick the tile's line stream.
    __builtin_prefetch((const void*)xin, 0, 0);
    __builtin_prefetch((const void*)(xin + (TT / 2) * Dv), 0, 0);

    // Load 4 weight rows (taps of 4 channels), transpose to per-tap vectors.
    v4f wr0 = wv[0], wr1 = wv[1], wr2 = wv[2], wr3 = wv[3];
    v4f k0 = {wr0.x, wr1.x, wr2.x, wr3.x};   // tap k=0 -> x[t-3]
    v4f k1 = {wr0.y, wr1.y, wr2.y, wr3.y};   // tap k=1 -> x[t-2]
    v4f k2 = {wr0.z, wr1.z, wr2.z, wr3.z};   // tap k=2 -> x[t-1]
    v4f k3 = {wr0.w, wr1.w, wr2.w, wr3.w};   // tap k=3 -> x[t]

    // Causal halo (t0==0 only for the first time tile; branch is block-uniform).
    v4f xm3, xm2, xm1;
    if (t0 >= 3) {
        xm3 = xin[-3 * Dv];     // imm offset -24KB
        xm2 = xin[-2 * Dv];
        xm1 = xin[-1 * Dv];
    } else {
        xm3 = (v4f)0.0f; xm2 = (v4f)0.0f; xm1 = (v4f)0.0f;
    }

#pragma unroll
    for (int i = 0; i < TT; ++i) {
        // Streamed, single-use data: non-temporal load (TH=NT).
        v4f xc = __builtin_nontemporal_load(xin + i * Dv);

        v4f y = k0 * xm3 + k1 * xm2 + k2 * xm1 + k3 * xc;

        // SiLU: y * sigmoid(y) = y * rcp(1 + exp(-y))
        v4f s;
        s.x = y.x * __builtin_amdgcn_rcpf(1.0f + __expf(-y.x));
        s.y = y.y * __builtin_amdgcn_rcpf(1.0f + __expf(-y.y));
        s.z = y.z * __builtin_amdgcn_rcpf(1.0f + __expf(-y.z));
        s.w = y.w * __builtin_amdgcn_rcpf(1.0f + __expf(-y.w));

        // Write-once output: non-temporal store, keep L2 for the x stream.
        __builtin_nontemporal_store(s, yout + i * Dv);

        xm3 = xm2; xm2 = xm1; xm1 = xc;
    }
}

extern "C" void kernel_launch(void* const* d_in, const int* in_sizes, int n_in,
                              void* d_out, int out_size, void* d_ws, size_t ws_size,
                              hipStream_t stream) {
    const float* x = (const float*)d_in[0];   // (B,T,D) fp32
    const float* w = (const float*)d_in[1];   // (D,4)   fp32
    float* out = (float*)d_out;

    // 4 * 256 * 512 = 524288 threads = 16384 wave32s
    constexpr long total = (long)cfg::B * cfg::tilesT * cfg::Dv;
    constexpr int block  = 256;
    constexpr unsigned grid = (unsigned)((total + block - 1) / block);

    dwconv4_silu_kernel<<<dim3(grid), dim3(block), 0, stream>>>(x, w, out);
}